// Transformer_Block_21827023798558
// MI455X (gfx1250) — compile-verified
//
#include <hip/hip_runtime.h>

typedef __attribute__((ext_vector_type(2))) float v2f;
typedef __attribute__((ext_vector_type(8))) float v8f;

// ---------------------------------------------------------------------------
// Zero-fill d_out (harness poisons it with 0xAA; atomics need a zeroed base).
// ---------------------------------------------------------------------------
__global__ __launch_bounds__(256) void zero_f4(float4* __restrict__ p, long n4) {
  long i = (long)blockIdx.x * blockDim.x + threadIdx.x;
  const long stride = (long)gridDim.x * blockDim.x;
  const float4 z = make_float4(0.f, 0.f, 0.f, 0.f);
  for (; i < n4; i += stride) p[i] = z;
}

// ---------------------------------------------------------------------------
// x3 = x * W3 (broadcast over C=128), float4 grid-stride. One pass over 51 MB;
// x3 then stays L2-resident and removes the W3 load stream from the hot loop.
// ---------------------------------------------------------------------------
__global__ __launch_bounds__(256)
void scale_w3(const float4* __restrict__ x, const float4* __restrict__ W3q,
              float4* __restrict__ x3, long n4) {
  long i = (long)blockIdx.x * blockDim.x + threadIdx.x;
  const long stride = (long)gridDim.x * blockDim.x;
  for (; i < n4; i += stride) {
    float4 v = x[i];
    const float4 w = W3q[i & 31];   // 128 channels = 32 float4s per row
    v.x *= w.x; v.y *= w.y; v.z *= w.z; v.w *= w.w;
    x3[i] = v;
  }
}

// Uniform lane-broadcast helpers: compile-time lane -> v_readlane_b32 (SGPR).
__device__ __forceinline__ int   bcast_i(int v, int lane) {
  return __builtin_amdgcn_readlane(v, lane);
}
__device__ __forceinline__ float bcast_f(float v, int lane) {
  return __builtin_bit_cast(
      float, __builtin_amdgcn_readlane(__builtin_bit_cast(int, v), lane));
}

// ---------------------------------------------------------------------------
// Per wave: 16 edges.
//  Phase 1: w[e] = dot(x3[src], x[dst]) over C=128 via 32 chained
//           V_WMMA_F32_16X16X4_F32; the 16 dot products are the diagonal of
//           the 16x16 fp32 accumulator. Loads are batched 8 chunks at a time
//           into explicit buffers so >=16 loads stay in flight per wave.
//  Phase 2: out[dst] += w * W2 * x[src]; edge scalars via v_readlane (SGPR),
//           rows gathered 4-at-a-time with global_load_b128 before the
//           atomic bursts.
// ---------------------------------------------------------------------------
template <bool PRE>
__global__ __launch_bounds__(256)
void edge_gnn_wmma(const float* __restrict__ xA,   // x3 if PRE, else x
                   const float* __restrict__ x,    // raw x
                   const int*   __restrict__ src,
                   const int*   __restrict__ dst,
                   const float* __restrict__ W2,
                   const float* __restrict__ W3,
                   float*       __restrict__ out,
                   const int E) {
  const int lane = (int)(threadIdx.x & 31u);
  const int wave = (int)((blockIdx.x * blockDim.x + threadIdx.x) >> 5);
  const long base = (long)wave * 16;
  if (base >= (long)E) return;                 // wave-uniform exit

  const int m = lane & 15;                     // lanes L and L+16 share edge L
  long e = base + m;
  if (e >= (long)E) e = (long)E - 1;           // clamp (tail); masked in phase 2
  const int sE = src[e];
  const int dE = dst[e];

  const float4 w2q = *(const float4*)(W2 + lane * 4);

  const float* __restrict__ xs = xA + (long)sE * 128;
  const float* __restrict__ xd = x  + (long)dE * 128;
  const int koff = (lane >> 4) << 1;           // 0 or 2

  v8f acc = {};                                 // 16x16 fp32 accumulator (D)
#pragma unroll
  for (int g = 0; g < 4; ++g) {                 // 4 groups x 8 chunks x K=4
    v2f abuf[8], bbuf[8], wbuf[8];
#pragma unroll
    for (int i = 0; i < 8; ++i) {               // batch: 16-24 loads in flight
      const int co = g * 32 + i * 4 + koff;
      abuf[i] = *(const v2f*)(xs + co);
      bbuf[i] = *(const v2f*)(xd + co);
      if (!PRE) wbuf[i] = *(const v2f*)(W3 + co);
    }
#pragma unroll
    for (int i = 0; i < 8; ++i) {
      const v2f a = PRE ? abuf[i] : (abuf[i] * wbuf[i]);
      acc = __builtin_amdgcn_wmma_f32_16x16x4_f32(
          false, a, false, bbuf[i], (short)0, acc, false, false);
    }
  }

  // Diagonal of D: w[m] at m<8 -> (vgpr m, lane m); m>=8 -> (vgpr m-8, lane m+16).
#pragma unroll
  for (int g = 0; g < 4; ++g) {                 // 4 groups x 4 edges
    float  wmv[4];
    int    s2v[4], d2v[4];
    float4 xv[4];
#pragma unroll
    for (int j = 0; j < 4; ++j) {               // gather rows first (overlap)
      const int mm = g * 4 + j;
      const int rr = (mm < 8) ? mm : mm - 8;
      const int ll = (mm < 8) ? mm : mm + 16;
      wmv[j] = bcast_f(acc[rr], ll);            // SGPR
      s2v[j] = bcast_i(sE, mm);                 // SGPR
      d2v[j] = bcast_i(dE, mm);                 // SGPR
      xv[j]  = *(const float4*)(x + (long)s2v[j] * 128 + lane * 4);
    }
#pragma unroll
    for (int j = 0; j < 4; ++j) {
      const int mm = g * 4 + j;
      if (base + mm < (long)E) {                // wave-uniform (scalar branch)
        float* __restrict__ orow = out + (long)d2v[j] * 128 + lane * 4;
        const float wm = wmv[j];
        atomicAdd(orow + 0, wm * w2q.x * xv[j].x);
        atomicAdd(orow + 1, wm * w2q.y * xv[j].y);
        atomicAdd(orow + 2, wm * w2q.z * xv[j].z);
        atomicAdd(orow + 3, wm * w2q.w * xv[j].w);
      }
    }
  }
}

extern "C" void kernel_launch(void* const* d_in, const int* in_sizes, int n_in,
                              void* d_out, int out_size, void* d_ws, size_t ws_size,
                              hipStream_t stream) {
  const float* x  = (const float*)d_in[0];   // (1, N, 128) fp32
  const int*   ei = (const int*)d_in[1];     // (2, E) int32
  const float* W2 = (const float*)d_in[2];   // (128,)
  const float* W3 = (const float*)d_in[3];   // (128,)
  float* out = (float*)d_out;                // (1, N, 128) fp32

  const int E = in_sizes[1] / 2;
  const long xElems = (long)in_sizes[0];     // N*128 floats

  // Zero output (grid-stride float4 fill).
  const long n4 = (long)out_size / 4;
  long zb = (n4 + 255) / 256;
  if (zb > 4096) zb = 4096;
  if (zb < 1) zb = 1;
  zero_f4<<<(int)zb, 256, 0, stream>>>((float4*)out, n4);

  // 16 edges per wave, 8 waves (256 threads) per block.
  const int waves  = (E + 15) / 16;
  const int blocks = (waves + 7) / 8;

  if (ws_size >= (size_t)xElems * sizeof(float)) {
    const long xq = xElems / 4;
    long sb = (xq + 255) / 256;
    if (sb > 4096) sb = 4096;
    if (sb < 1) sb = 1;
    scale_w3<<<(int)sb, 256, 0, stream>>>((const float4*)x, (const float4*)W3,
                                          (float4*)d_ws, xq);
    edge_gnn_wmma<true><<<blocks, 256, 0, stream>>>(
        (const float*)d_ws, x, ei, ei + E, W2, W3, out, E);
  } else {
    edge_gnn_wmma<false><<<blocks, 256, 0, stream>>>(
        x, x, ei, ei + E, W2, W3, out, E);
  }
}